// PairList_26938034880563
// MI455X (gfx1250) — compile-verified
//
#include <hip/hip_runtime.h>
#include <hip/hip_bf16.h>
#include <math.h>

// CDNA5 / gfx1250 pair-list kernel.
// One wave32 owns a 16-row i-strip and GROUP=8 consecutive 16-col j-tiles.
// Squared distances come from the f32 matrix core (v_wmma_f32_16x16x4_f32)
// via the Gram augmentation:
//   d^2 = (-2x_i,-2y_i,-2z_i,1) . (x_j,y_j,z_j,|r_j|^2) + |r_i|^2
// Row-side state (coords, norms, A operand, broadcasts, triangular row bases)
// is computed once per wave; all 8 j-tile coord loads are issued up front;
// every output section is addressed as SGPR-base + 32-bit index (GVS
// scale_offset stores). Workload is store-bandwidth-bound (~470 MB out).

typedef __attribute__((ext_vector_type(2))) float v2f;
typedef __attribute__((ext_vector_type(8))) float v8f;

#define NATOMS 2048
#define TILE   16
#define GROUP  8          // j-tiles per wave
#define CUTOFF 5.0f

struct F3 { float x, y, z; };   // 12-byte r_ij record -> b96 scale_offset store

__global__ __launch_bounds__(32)
void PairList_gfx1250_kernel(const float* __restrict__ R,
                             float* __restrict__ outI,
                             float* __restrict__ outJ,
                             float* __restrict__ outD,
                             F3*    __restrict__ outR,
                             float* __restrict__ outM,
                             long long P)     // pairs per molecule
{
    const int ti      = blockIdx.x;             // i tile block
    const int tjStart = blockIdx.y * GROUP;     // first j tile of this group
    if (tjStart + GROUP - 1 < ti) return;       // group entirely below diagonal
    const int b = blockIdx.z;

    const int lane   = threadIdx.x;             // wave32
    const int laneLo = lane & 15;
    const int hi     = lane >> 4;               // 0: K=0,1 half ; 1: K=2,3 half

    const float* __restrict__ Rb = R + (size_t)b * NATOMS * 3;
    const int i0 = ti * TILE;

    // Bias section bases by this molecule's pair offset (uniform, stays SGPR).
    const long long gBase = (long long)b * P;
    outI += gBase; outJ += gBase; outD += gBase; outM += gBase; outR += gBase;

    // ---- Row-side (A) coords ----
    const float xi = Rb[(i0 + laneLo) * 3 + 0];
    const float yi = Rb[(i0 + laneLo) * 3 + 1];
    const float zi = Rb[(i0 + laneLo) * 3 + 2];

    // ---- Preload ALL col-side coords for the 8 j-tiles (one load clause) ----
    float xjA[GROUP], yjA[GROUP], zjA[GROUP];
#pragma unroll
    for (int t = 0; t < GROUP; ++t) {
        const int j = (tjStart + t) * TILE + laneLo;
        xjA[t] = Rb[j * 3 + 0];
        yjA[t] = Rb[j * 3 + 1];
        zjA[t] = Rb[j * 3 + 2];
    }

    const float ni = xi * xi + yi * yi + zi * zi;   // |r_i|^2

    // A operand (16x4 f32): lanes 0-15 -> K0=-2x_i, K1=-2y_i
    //                       lanes 16-31 -> K2=-2z_i, K3=1
    v2f a;
    a.x = hi ? (-2.0f * zi) : (-2.0f * xi);
    a.y = hi ? 1.0f         : (-2.0f * yi);

    const int fioff = b * NATOMS;

    // Per-accumulator-row constants (rows m = v + 8*hi), shared by all tiles.
    // (ds_bpermute broadcasts overlap the in-flight global loads above.)
    float rxi[8], ryi[8], rzi[8], nrmi[8], fi[8];
    int   iRow[8], pBase[8];
#pragma unroll
    for (int v = 0; v < 8; ++v) {
        const int m = v + 8 * hi;
        const int i = i0 + m;
        rxi[v]  = __shfl(xi, m, 32);
        ryi[v]  = __shfl(yi, m, 32);
        rzi[v]  = __shfl(zi, m, 32);
        nrmi[v] = __shfl(ni, m, 32);
        iRow[v] = i;
        fi[v]   = (float)(fioff + i);
        // triu flat index: p(i,j) = pBase[v] + j   (32-bit, p < 2^21)
        pBase[v] = i * (NATOMS - 1) - ((i * (i - 1)) >> 1) - i - 1;
    }

    // ---- Process the 8 j-tiles (fully unrolled, coords already resident) ----
#pragma unroll
    for (int t = 0; t < GROUP; ++t) {
        const int tj = tjStart + t;
        if (tj < ti) continue;                  // below diagonal
        const int j  = tj * TILE + laneLo;      // this lane's column

        const float xj = xjA[t];
        const float yj = yjA[t];
        const float zj = zjA[t];
        const float nj = xj * xj + yj * yj + zj * zj;   // |r_j|^2

        // B operand (4x16 f32): lanes 0-15 -> K0=x_j, K1=y_j
        //                       lanes 16-31 -> K2=z_j, K3=|r_j|^2
        v2f bmat;
        bmat.x = hi ? zj : xj;
        bmat.y = hi ? nj : yj;

        v8f c = {};
        // D[m][n] = -2 r_i.r_j + |r_j|^2   (add |r_i|^2 per element below)
        v8f d2 = __builtin_amdgcn_wmma_f32_16x16x4_f32(
            /*neg_a=*/false, a, /*neg_b=*/false, bmat,
            /*c_mod=*/(short)0, c, /*reuse_a=*/false, /*reuse_b=*/false);

        const float fj = (float)(fioff + j);

#define PAIR_BODY(MASKED)                                                     \
        _Pragma("unroll")                                                     \
        for (int v = 0; v < 8; ++v) {                                         \
            const float rx  = rxi[v] - xj;                                    \
            const float ry  = ryi[v] - yj;                                    \
            const float rz  = rzi[v] - zj;                                    \
            const float dsq = fmaxf(d2[v] + nrmi[v], 0.0f);                   \
            const float d   = __builtin_amdgcn_sqrtf(dsq);                    \
            if (!MASKED || (j > iRow[v])) {                                   \
                const int p = pBase[v] + j;      /* triu flat pair index */   \
                outI[p] = fi[v];                 /* atom_index12[0] */        \
                outJ[p] = fj;                    /* atom_index12[1] */        \
                outD[p] = d;                     /* d_ij            */        \
                F3 r3; r3.x = rx; r3.y = ry; r3.z = rz;                       \
                outR[p] = r3;                    /* r_ij (b96)      */        \
                outM[p] = (dsq <= CUTOFF * CUTOFF) ? 1.0f : 0.0f;             \
            }                                                                 \
        }

        if (tj > ti) {
            PAIR_BODY(false)   // strictly above diagonal: all 256 pairs valid
        } else {
            PAIR_BODY(true)    // diagonal tile: mask j > i
        }
#undef PAIR_BODY
    }
}

extern "C" void kernel_launch(void* const* d_in, const int* in_sizes, int n_in,
                              void* d_out, int out_size, void* d_ws, size_t ws_size,
                              hipStream_t stream) {
    (void)n_in; (void)out_size; (void)d_ws; (void)ws_size;
    const float* R = (const float*)d_in[0];
    const int B = in_sizes[0] / (3 * NATOMS);          // = 8 for the reference
    const long long P  = (long long)NATOMS * (NATOMS - 1) / 2;
    const long long BP = (long long)B * P;

    float* out  = (float*)d_out;
    float* outI = out;                  // atom_index12[0]   [BP]
    float* outJ = out + BP;             // atom_index12[1]   [BP]
    float* outD = out + 2 * BP;         // d_ij              [BP]
    F3*    outR = (F3*)(out + 3 * BP);  // r_ij              [BP][3]
    float* outM = out + 6 * BP;         // in_cutoff         [BP]

    dim3 grid(NATOMS / TILE, NATOMS / (TILE * GROUP), B);  // 128 x 16 x 8
    dim3 block(32, 1, 1);                                  // one wave32
    PairList_gfx1250_kernel<<<grid, block, 0, stream>>>(
        R, outI, outJ, outD, outR, outM, P);
}